// MoH_MDTA_Attention_TaskMI_85315230368365
// MI455X (gfx1250) — compile-verified
//
#include <hip/hip_runtime.h>
#include <hip/hip_bf16.h>

// ---------------- problem constants (from reference) ----------------
#define BATCH   8
#define DIM     192
#define CH3     (3 * DIM)      // 576
#define HEADS   8
#define HD      (DIM / HEADS)  // 24
#define TOPK    2
#define HH      128
#define WW      128
#define NPIX    (HH * WW)      // 16384
#define SPLIT   16             // split-K blocks per (b,h) in attn_stats
#define NT      128            // n-chunk staged in LDS in attn_stats
#define SCALE_ATTN 0.20412414523193154f   // 24^-0.5
#define EPS_F32 1.1920929e-07f            // finfo(float32).eps

typedef __attribute__((ext_vector_type(2))) float v2f;
typedef __attribute__((ext_vector_type(8))) float v8f;

// =====================================================================
// GEMM via V_WMMA_F32_16X16X4_F32:  Y[b] = W x X[b]
//   W: [M,K] row-major,  X: [B][K,N] row-major,  Y: [B][M,N]
// Each wave owns one 16-wide column strip (n0) and MB row tiles,
// reusing the B fragment across the MB tiles each K-step.
// ISA layouts (CDNA5 ISA §7.12.2):
//   A 16x4 : lane l<16 -> M=l, VGPR{0,1}=K{0,1}; lanes 16-31 -> K{2,3}
//   B 4x16 : lane l<16 -> N=l, VGPR{0,1}=K{0,1}; lanes 16-31 -> K{2,3}
//   D 16x16: VGPR r, lane l<16 -> (M=r,   N=l); lanes 16-31 -> (M=8+r, N=l)
// =====================================================================
template <int MB>
__global__ void gemm_wmma_f32(const float* __restrict__ Wm,
                              const float* __restrict__ X,
                              float* __restrict__ Y,
                              int M, int K, int Ncols)
{
    const int lane = threadIdx.x & 31;
    const int wave = threadIdx.x >> 5;          // 4 waves per block
    const int half = lane >> 4;                 // 0: K0/K1 , 1: K2/K3
    const int l    = lane & 15;
    const int b    = blockIdx.z;
    const int n0   = (blockIdx.x * 4 + wave) * 16;
    const int m0   = blockIdx.y * (MB * 16);

    const float* __restrict__ Xb = X + (size_t)b * (size_t)K * (size_t)Ncols;
    float* __restrict__       Yb = Y + (size_t)b * (size_t)M * (size_t)Ncols;

    v8f acc[MB];
#pragma unroll
    for (int i = 0; i < MB; ++i) { v8f z = {}; acc[i] = z; }

    for (int k = 0; k < K; k += 4) {
        v2f bfrag;
        bfrag.x = Xb[(size_t)(k + 2 * half)     * Ncols + n0 + l];
        bfrag.y = Xb[(size_t)(k + 2 * half + 1) * Ncols + n0 + l];
#pragma unroll
        for (int i = 0; i < MB; ++i) {
            const float* wr = Wm + (size_t)(m0 + i * 16 + l) * K + k + 2 * half;
            v2f afrag;
            afrag.x = wr[0];
            afrag.y = wr[1];
            acc[i] = __builtin_amdgcn_wmma_f32_16x16x4_f32(
                         false, afrag, false, bfrag, (short)0, acc[i], false, false);
        }
    }

#pragma unroll
    for (int i = 0; i < MB; ++i) {
        const int mrow = m0 + i * 16 + half * 8;
#pragma unroll
        for (int r = 0; r < 8; ++r)
            Yb[(size_t)(mrow + r) * Ncols + n0 + l] = acc[i][r];
    }
}

// =====================================================================
// Router: per-pixel 8-way softmax -> top-2 -> renormalized gates
// gates[b,h,n] (raw; *TOPK applied in attn_apply)
// =====================================================================
__global__ void router_gates(const float* __restrict__ x,
                             const float* __restrict__ rw_main,
                             const float* __restrict__ rw_aux,
                             const int* __restrict__ task_id,
                             float* __restrict__ gates)
{
    __shared__ float w[HEADS * DIM];
    const float* rw = (task_id[0] == 0) ? rw_main : rw_aux;
    for (int i = threadIdx.x; i < HEADS * DIM; i += blockDim.x) w[i] = rw[i];
    __syncthreads();

    const int b = blockIdx.y;
    const int n = blockIdx.x * blockDim.x + threadIdx.x;
    const float* xb = x + (size_t)b * DIM * NPIX + n;

    float lg[HEADS];
#pragma unroll
    for (int h = 0; h < HEADS; ++h) lg[h] = 0.f;

    for (int c = 0; c < DIM; ++c) {
        float xv = xb[(size_t)c * NPIX];
#pragma unroll
        for (int h = 0; h < HEADS; ++h) lg[h] += xv * w[h * DIM + c];
    }

    float m = lg[0];
#pragma unroll
    for (int h = 1; h < HEADS; ++h) m = fmaxf(m, lg[h]);
    float p[HEADS], s = 0.f;
#pragma unroll
    for (int h = 0; h < HEADS; ++h) { p[h] = expf(lg[h] - m); s += p[h]; }
    float inv = 1.f / s;
#pragma unroll
    for (int h = 0; h < HEADS; ++h) p[h] *= inv;

    int i1 = 0;
#pragma unroll
    for (int h = 1; h < HEADS; ++h) if (p[h] > p[i1]) i1 = h;
    int i2 = (i1 == 0) ? 1 : 0;
#pragma unroll
    for (int h = 0; h < HEADS; ++h)
        if (h != i1 && h != i2 && p[h] > p[i2]) i2 = h;

    float denom = fmaxf(p[i1] + p[i2], EPS_F32);
#pragma unroll
    for (int h = 0; h < HEADS; ++h) {
        float g = (h == i1 || h == i2) ? p[h] / denom : 0.f;
        gates[((size_t)b * HEADS + h) * NPIX + n] = g;
    }
}

// =====================================================================
// Depthwise 3x3, padding 1, 576 channels (cross-correlation, NN conv)
// =====================================================================
__global__ void dwconv3x3(const float* __restrict__ in,
                          const float* __restrict__ wgt,
                          float* __restrict__ out)
{
    const int ch  = blockIdx.y;
    const int b   = blockIdx.z;
    const int pix = blockIdx.x * blockDim.x + threadIdx.x;
    const int y   = pix >> 7;       // /128
    const int xw  = pix & 127;

    const float* ip = in + ((size_t)b * CH3 + ch) * (size_t)NPIX;
    const float* wp = wgt + (size_t)ch * 9;

    float s = 0.f;
#pragma unroll
    for (int dy = -1; dy <= 1; ++dy) {
#pragma unroll
        for (int dx = -1; dx <= 1; ++dx) {
            int yy = y + dy, xx = xw + dx;
            if (yy >= 0 && yy < HH && xx >= 0 && xx < WW)
                s += wp[(dy + 1) * 3 + (dx + 1)] * ip[yy * WW + xx];
        }
    }
    out[((size_t)b * CH3 + ch) * (size_t)NPIX + pix] = s;
}

// =====================================================================
// attn_stats: split-K accumulation of S[c,d]=sum_n q_c k_d, qq, kk
// sbuf layout per (b,h): [0..575]=S, [576..599]=qq, [600..623]=kk
// =====================================================================
__global__ void attn_stats(const float* __restrict__ qkv2,
                           float* __restrict__ sbuf)
{
    __shared__ float qs[HD * NT];
    __shared__ float ks[HD * NT];
    const int tid = threadIdx.x;
    const int h = blockIdx.y, b = blockIdx.z;
    const int nbeg = blockIdx.x * (NPIX / SPLIT);

    const size_t qbase = ((size_t)b * CH3 + h * HD) * (size_t)NPIX;
    const size_t kbase = ((size_t)b * CH3 + DIM + h * HD) * (size_t)NPIX;

    const int p0 = tid, p1 = tid + 256, p2 = tid + 512;
    const int c0 = p0 / HD, d0 = p0 % HD;
    const int c1 = p1 / HD, d1 = p1 % HD;
    const int c2 = (tid < 64) ? (p2 / HD) : 0;
    const int d2 = (tid < 64) ? (p2 % HD) : 0;

    float a0 = 0.f, a1 = 0.f, a2 = 0.f, qqa = 0.f, kka = 0.f;

    for (int n0 = nbeg; n0 < nbeg + NPIX / SPLIT; n0 += NT) {
        __syncthreads();
        for (int i = tid; i < HD * NT; i += 256) {
            int c = i / NT, j = i & (NT - 1);
            qs[i] = qkv2[qbase + (size_t)c * NPIX + n0 + j];
            ks[i] = qkv2[kbase + (size_t)c * NPIX + n0 + j];
        }
        __syncthreads();
        for (int j = 0; j < NT; ++j) {
            a0 += qs[c0 * NT + j] * ks[d0 * NT + j];
            a1 += qs[c1 * NT + j] * ks[d1 * NT + j];
            a2 += qs[c2 * NT + j] * ks[d2 * NT + j];
        }
        if (tid < HD) {
            for (int j = 0; j < NT; ++j) { float v = qs[tid * NT + j]; qqa += v * v; }
        } else if (tid < 2 * HD) {
            for (int j = 0; j < NT; ++j) { float v = ks[(tid - HD) * NT + j]; kka += v * v; }
        }
    }

    const size_t base = (size_t)(b * HEADS + h) * 624;
    atomicAdd(&sbuf[base + p0], a0);
    atomicAdd(&sbuf[base + p1], a1);
    if (tid < 64)          atomicAdd(&sbuf[base + p2], a2);
    if (tid < HD)          atomicAdd(&sbuf[base + 576 + tid], qqa);
    else if (tid < 2 * HD) atomicAdd(&sbuf[base + 600 + (tid - HD)], kka);
}

// =====================================================================
// attn_softmax: normalize by L2 norms (eps 1e-12), *scale, row softmax
// =====================================================================
__global__ void attn_softmax(const float* __restrict__ sbuf,
                             float* __restrict__ attnbuf)
{
    const int bh = blockIdx.x;
    const int c  = threadIdx.x;
    if (c >= HD) return;
    const float* sb = sbuf + (size_t)bh * 624;

    const float qn = fmaxf(sqrtf(sb[576 + c]), 1e-12f);
    float row[HD];
#pragma unroll
    for (int d = 0; d < HD; ++d) {
        float kn = fmaxf(sqrtf(sb[600 + d]), 1e-12f);
        row[d] = sb[c * HD + d] / (qn * kn) * SCALE_ATTN;
    }
    float m = row[0];
#pragma unroll
    for (int d = 1; d < HD; ++d) m = fmaxf(m, row[d]);
    float s = 0.f;
#pragma unroll
    for (int d = 0; d < HD; ++d) { row[d] = expf(row[d] - m); s += row[d]; }
    float inv = 1.f / s;
#pragma unroll
    for (int d = 0; d < HD; ++d)
        attnbuf[(size_t)bh * (HD * HD) + c * HD + d] = row[d] * inv;
}

// =====================================================================
// attn_apply: out[c,n] = (sum_d attn[c,d] v[d,n]) * gate[b,h,n] * TOPK
// =====================================================================
__global__ void attn_apply(const float* __restrict__ qkv2,
                           const float* __restrict__ attnbuf,
                           const float* __restrict__ gates,
                           float* __restrict__ gout)
{
    __shared__ float a[HD * HD];
    const int tid = threadIdx.x;
    const int h = blockIdx.y, b = blockIdx.z;
    const int bh = b * HEADS + h;
    for (int i = tid; i < HD * HD; i += 256) a[i] = attnbuf[(size_t)bh * (HD * HD) + i];
    __syncthreads();

    const int n = blockIdx.x * 256 + tid;
    const float g = gates[(size_t)bh * NPIX + n] * (float)TOPK;

    const float* vp = qkv2 + ((size_t)b * CH3 + 2 * DIM + h * HD) * (size_t)NPIX + n;
    float vv[HD];
#pragma unroll
    for (int d = 0; d < HD; ++d) vv[d] = vp[(size_t)d * NPIX];

    float* op = gout + ((size_t)b * DIM + h * HD) * (size_t)NPIX + n;
#pragma unroll
    for (int c = 0; c < HD; ++c) {
        float s = 0.f;
#pragma unroll
        for (int d = 0; d < HD; ++d) s += a[c * HD + d] * vv[d];
        op[(size_t)c * NPIX] = s * g;
    }
}

// =====================================================================
// Launcher.  Workspace layout (bytes):
//   [0, S1)            qkv1  (1x1 conv out)   -> later aliased by gout
//   [S1, 2*S1)         qkv2  (depthwise out)
//   [2*S1, +4MB)       gates [B,HEADS,N]
//   next 160KB         sbuf  (S/qq/kk accumulators)
//   next 148KB         attnbuf
// Peak ~608.5 MB.
// =====================================================================
extern "C" void kernel_launch(void* const* d_in, const int* in_sizes, int n_in,
                              void* d_out, int out_size, void* d_ws, size_t ws_size,
                              hipStream_t stream)
{
    const float* x       = (const float*)d_in[0];
    const float* qkv_w   = (const float*)d_in[1];
    const float* dw_w    = (const float*)d_in[2];
    const float* proj_w  = (const float*)d_in[3];
    const float* rw_main = (const float*)d_in[4];
    const float* rw_aux  = (const float*)d_in[5];
    const int*   task_id = (const int*)d_in[6];
    float* out = (float*)d_out;

    char* ws = (char*)d_ws;
    const size_t S1 = (size_t)BATCH * CH3 * NPIX * sizeof(float);        // 302 MB
    const size_t GB = (size_t)BATCH * HEADS * NPIX * sizeof(float);      // 4 MB
    const size_t SB = (size_t)BATCH * HEADS * 624 * sizeof(float);       // 160 KB

    float* qkv1    = (float*)(ws);
    float* qkv2    = (float*)(ws + S1);
    float* gates   = (float*)(ws + 2 * S1);
    float* sbuf    = (float*)(ws + 2 * S1 + GB);
    float* attnbuf = (float*)(ws + 2 * S1 + GB + SB);
    float* gout    = qkv1;   // qkv1 dead after dwconv

    hipMemsetAsync(sbuf, 0, SB, stream);

    router_gates<<<dim3(NPIX / 256, BATCH), 256, 0, stream>>>(
        x, rw_main, rw_aux, task_id, gates);

    // QKV 1x1 conv: [576,192] x [192,16384] per batch
    gemm_wmma_f32<6><<<dim3(NPIX / 64, CH3 / 96, BATCH), 128, 0, stream>>>(
        qkv_w, x, qkv1, CH3, DIM, NPIX);

    dwconv3x3<<<dim3(NPIX / 256, CH3, BATCH), 256, 0, stream>>>(
        qkv1, dw_w, qkv2);

    attn_stats<<<dim3(SPLIT, HEADS, BATCH), 256, 0, stream>>>(qkv2, sbuf);
    attn_softmax<<<dim3(BATCH * HEADS), 32, 0, stream>>>(sbuf, attnbuf);
    attn_apply<<<dim3(NPIX / 256, HEADS, BATCH), 256, 0, stream>>>(
        qkv2, attnbuf, gates, gout);

    // proj 1x1 conv: [192,192] x [192,16384] per batch -> d_out
    gemm_wmma_f32<6><<<dim3(NPIX / 64, DIM / 96, BATCH), 128, 0, stream>>>(
        proj_w, gout, out, DIM, DIM, NPIX);
}